// ChamferLoss_17239998726835
// MI455X (gfx1250) — compile-verified
//
#include <hip/hip_runtime.h>
#include <hip/hip_bf16.h>

typedef float v2f __attribute__((ext_vector_type(2)));
typedef float v8f __attribute__((ext_vector_type(8)));
typedef int   i32x4 __attribute__((vector_size(16)));   // matches builtin param type

#define B_ 4
#define N_ 8192
#define M_ 8192
#define D_ 64

#define MT 64           // M-tile (columns staged in LDS per iteration)
#define LDS_STRIDE 68   // 64 + 4 floats pad: keeps 16B alignment, avoids bank conflicts

// ---- CDNA5 async global->LDS staging (ASYNCcnt path), with safe fallback ----
#if defined(__has_builtin)
#  if __has_builtin(__builtin_amdgcn_global_load_async_to_lds_b128)
#    define HAVE_ASYNC_LDS 1
#  endif
#endif
#ifndef HAVE_ASYNC_LDS
#  define HAVE_ASYNC_LDS 0
#endif

typedef __attribute__((address_space(1))) i32x4 g_i32x4;
typedef __attribute__((address_space(3))) i32x4 l_i32x4;

__device__ __forceinline__ void wait_asynccnt0() {
#if defined(__has_builtin)
#  if __has_builtin(__builtin_amdgcn_s_wait_asynccnt)
    __builtin_amdgcn_s_wait_asynccnt(0);
    return;
#  else
    asm volatile("s_wait_asynccnt 0x0" ::: "memory");
    return;
#  endif
#else
    asm volatile("s_wait_asynccnt 0x0" ::: "memory");
#endif
}

// ---- order-preserving float <-> uint key (handles negative round-off safely) ----
__device__ __forceinline__ unsigned fkey(float x) {
    unsigned b = __float_as_uint(x);
    return (b & 0x80000000u) ? ~b : (b | 0x80000000u);
}
__device__ __forceinline__ float funkey(unsigned u) {
    unsigned b = (u & 0x80000000u) ? (u & 0x7FFFFFFFu) : ~u;
    return __uint_as_float(b);
}

// ------------------------------------------------------------------
// Kernel 0: per-row squared norms of f (B*N rows) and f_ (B*M rows),
//           written contiguously into sqn[]; also init colkey[] = +max.
// ------------------------------------------------------------------
__global__ __launch_bounds__(256) void norms_init_kernel(
    const float* __restrict__ f, const float* __restrict__ f_,
    float* __restrict__ sqn, unsigned* __restrict__ colkey)
{
    const int BN = B_ * N_, BM = B_ * M_;
    int tid = blockIdx.x * blockDim.x + threadIdx.x;
    if (tid < BN + BM) {
        const float* p = (tid < BN) ? (f + (size_t)tid * D_)
                                    : (f_ + (size_t)(tid - BN) * D_);
        const float4* p4 = (const float4*)p;
        float s = 0.f;
        #pragma unroll
        for (int j = 0; j < D_ / 4; ++j) {
            float4 v = p4[j];
            s += v.x * v.x + v.y * v.y + v.z * v.z + v.w * v.w;
        }
        sqn[tid] = s;
    }
    if (tid < BM) colkey[tid] = 0xFFFFFFFFu;
}

// ------------------------------------------------------------------
// Main kernel: 128 threads = 4 wave32s. Block owns 64 rows of one
// batch; each wave owns 16 rows (full K=64 A-fragments in registers).
// Loop over M in 64-col LDS tiles; 64 x v_wmma_f32_16x16x4_f32 per
// tile per wave; row mins kept in regs, col mins folded via LDS.
// ------------------------------------------------------------------
__global__ __launch_bounds__(128) void chamfer_tiles_kernel(
    const float* __restrict__ f, const float* __restrict__ f_,
    const float* __restrict__ sqn_f, const float* __restrict__ sqn_f_,
    float* __restrict__ rowmin, unsigned* __restrict__ colkey)
{
    __shared__ float    tile[MT * LDS_STRIDE];
    __shared__ float    cn_s[MT];
    __shared__ unsigned cm_s[MT];

    const int strips = N_ / 64;
    const int strip  = blockIdx.x % strips;
    const int b      = blockIdx.x / strips;
    const int n0     = strip * 64;
    const int tid    = threadIdx.x;
    const int wave   = tid >> 5;
    const int lane   = tid & 31;
    const int l16    = lane & 15;
    const int hi     = lane >> 4;

    const float* fb  = f  + (size_t)b * N_ * D_;
    const float* f_b = f_ + (size_t)b * M_ * D_;

    // ---- preload A fragments: 16 rows x K=64 per wave ----
    // A 16x4 f32 layout: lanes 0-15 hold rows, VGPR pair = (K, K+1) for lo
    // lanes and (K+2, K+3) for hi lanes.
    const int arow = n0 + wave * 16 + l16;
    v2f a[16];
    {
        const float* ap = fb + (size_t)arow * D_ + 2 * hi;
        #pragma unroll
        for (int j = 0; j < 16; ++j) a[j] = *(const v2f*)(ap + 4 * j);
    }
    // row norms for the C/D layout: VGPR i <-> row i + 8*hi
    float rn[8];
    #pragma unroll
    for (int i = 0; i < 8; ++i)
        rn[i] = sqn_f[(size_t)b * N_ + n0 + wave * 16 + i + 8 * hi];

    float r[8];
    #pragma unroll
    for (int i = 0; i < 8; ++i) r[i] = 3.0e38f;

    const int ldsoff = l16 * LDS_STRIDE + 2 * hi;

    for (int m0 = 0; m0 < M_; m0 += MT) {
        // ---- stage f_ tile (64 rows x 64 floats) into LDS ----
        if (tid < MT) {
            cn_s[tid] = sqn_f_[(size_t)b * M_ + m0 + tid];
            cm_s[tid] = 0xFFFFFFFFu;
        }
#if HAVE_ASYNC_LDS
        #pragma unroll
        for (int j = 0; j < 8; ++j) {
            int idx = tid + j * 128;     // float4 index, 1024 total
            int m   = idx >> 4;
            int c4  = idx & 15;
            __builtin_amdgcn_global_load_async_to_lds_b128(
                (g_i32x4*)(f_b + (size_t)(m0 + m) * D_ + c4 * 4),
                (l_i32x4*)(&tile[m * LDS_STRIDE + c4 * 4]),
                0, 0);
        }
        wait_asynccnt0();
#else
        #pragma unroll
        for (int j = 0; j < 8; ++j) {
            int idx = tid + j * 128;     // float4 index, 1024 total
            int m   = idx >> 4;
            int c4  = idx & 15;
            float4 v = *(const float4*)(f_b + (size_t)(m0 + m) * D_ + c4 * 4);
            *(float4*)&tile[m * LDS_STRIDE + c4 * 4] = v;
        }
#endif
        __syncthreads();

        // ---- 4 column sub-tiles of 16: K=64 via 16 WMMA steps each ----
        #pragma unroll
        for (int sub = 0; sub < 4; ++sub) {
            v8f c = {};
            const float* bp = &tile[sub * 16 * LDS_STRIDE + ldsoff];
            #pragma unroll
            for (int j = 0; j < 16; ++j) {
                v2f bb = *(const v2f*)(bp + 4 * j);
                c = __builtin_amdgcn_wmma_f32_16x16x4_f32(
                        false, a[j], false, bb, (short)0, c, false, false);
            }
            const float cn = cn_s[sub * 16 + l16];
            float tmin = 3.0e38f;
            #pragma unroll
            for (int i = 0; i < 8; ++i) {
                float d = fmaf(c[i], -2.0f, rn[i] + cn);   // ||x||^2+||y||^2-2xy
                r[i]  = fminf(r[i], d);
                tmin  = fminf(tmin, d);
            }
            // fold the two row-halves, then combine across waves in LDS
            tmin = fminf(tmin, __shfl_xor(tmin, 16, 32));
            if (lane < 16) atomicMin(&cm_s[sub * 16 + lane], fkey(tmin));
        }
        __syncthreads();

        // ---- flush this tile's column mins to global ----
        if (tid < MT)
            atomicMin(&colkey[(size_t)b * M_ + m0 + tid], cm_s[tid]);
        // (next-iteration LDS writes are safe: flush reader == re-init writer)
    }

    // ---- finalize row mins: reduce across the 16 lanes of each half ----
    #pragma unroll
    for (int i = 0; i < 8; ++i) {
        float v = r[i];
        v = fminf(v, __shfl_xor(v, 1, 32));
        v = fminf(v, __shfl_xor(v, 2, 32));
        v = fminf(v, __shfl_xor(v, 4, 32));
        v = fminf(v, __shfl_xor(v, 8, 32));
        if (l16 == 0)
            rowmin[(size_t)b * N_ + n0 + wave * 16 + i + 8 * hi] = v;
    }
}

// ------------------------------------------------------------------
// Final reduction: mean_b( mean_n rowmin + mean_m colmin ) -> scalar
// ------------------------------------------------------------------
__global__ __launch_bounds__(256) void finalize_kernel(
    const float* __restrict__ rowmin, const unsigned* __restrict__ colkey,
    float* __restrict__ out)
{
    __shared__ float red[256];
    const int tid = threadIdx.x;
    const float wr = 1.0f / ((float)B_ * (float)N_);
    const float wc = 1.0f / ((float)B_ * (float)M_);
    float s = 0.f;
    for (int i = tid; i < B_ * N_; i += 256) s += rowmin[i] * wr;
    for (int i = tid; i < B_ * M_; i += 256) s += funkey(colkey[i]) * wc;
    red[tid] = s;
    __syncthreads();
    for (int off = 128; off > 0; off >>= 1) {
        if (tid < off) red[tid] += red[tid + off];
        __syncthreads();
    }
    if (tid == 0) out[0] = red[0];
}

// ------------------------------------------------------------------
extern "C" void kernel_launch(void* const* d_in, const int* in_sizes, int n_in,
                              void* d_out, int out_size, void* d_ws, size_t ws_size,
                              hipStream_t stream)
{
    (void)in_sizes; (void)n_in; (void)out_size; (void)ws_size;
    const float* f  = (const float*)d_in[0];
    const float* f_ = (const float*)d_in[1];

    const int BN = B_ * N_, BM = B_ * M_;
    float*    ws      = (float*)d_ws;
    float*    sqn     = ws;                      // [0, BN+BM)  : f norms then f_ norms
    float*    sqn_f   = sqn;
    float*    sqn_f_  = sqn + BN;
    float*    rowmin  = ws + BN + BM;            // [.., +BN)
    unsigned* colkey  = (unsigned*)(ws + 2 * BN + BM);  // [.., +BM)
    // total scratch: 2*(BN+BM)*4 = 512 KB

    // 1) norms + colkey init
    {
        int total = BN + BM;
        norms_init_kernel<<<(total + 255) / 256, 256, 0, stream>>>(f, f_, sqn, colkey);
    }
    // 2) tiled WMMA distance + running mins
    {
        dim3 grid(B_ * (N_ / 64));
        chamfer_tiles_kernel<<<grid, 128, 0, stream>>>(f, f_, sqn_f, sqn_f_,
                                                       rowmin, colkey);
    }
    // 3) scalar reduction
    finalize_kernel<<<1, 256, 0, stream>>>(rowmin, colkey, (float*)d_out);
}